// NetworkRNNCell_16389595201753
// MI455X (gfx1250) — compile-verified
//
#include <hip/hip_runtime.h>
#include <hip/hip_bf16.h>
#include <math.h>

// ---------------------------------------------------------------------------
// NetworkRNNCell on MI455X (gfx1250).
//
// Memory-bound: streaming W (20 x 4096 x 4096 fp32 = 1.34 GB) at 23.3 TB/s
// sets a ~58 us floor. fp32 end-to-end (matches reference precision);
// V_WMMA_F32_16X16X4_F32 is the GEMV accumulation engine: A-matrix row M=0
// carries the pre-synaptic rate vector (K=4 per step), B carries 4x16 tiles
// of W, D row 0 (VGPR 0, lanes 0..15) holds the 16 partial dot products.
// Deterministic: split-K partials + fixed-order reductions, no float atomics.
// ---------------------------------------------------------------------------

typedef float v2f __attribute__((ext_vector_type(2)));
typedef float v8f __attribute__((ext_vector_type(8)));

namespace {
constexpr int   NN    = 4096;   // units per population
constexpr int   P     = 4;      // dynamic populations
constexpr int   S     = 20;     // synaptic projections
constexpr int   JT    = 8;      // 16-wide j-tiles per wave (128 columns)
constexpr int   JB2   = NN / (16 * JT);   // 32 column blocks
constexpr int   IB    = 8;      // split-K i-chunks
constexpr int   ICH   = NN / IB;          // 512 rows per chunk
constexpr float DT    = 0.1f;
constexpr float TAU_M = 10.0f;
constexpr float TAU_S = 5.0f;
constexpr float IN_FREQ = 0.008f;
constexpr float TWO_PI  = 6.28318530717958647692f;

// workspace layout (floats); total ~2.7 MB
constexpr size_t WS_RATES = 0;                                // [5][NN]
constexpr size_t WS_DRIVE = WS_RATES + 5 * NN;                // [IB][S][NN]
constexpr size_t WS_ERR   = WS_DRIVE + (size_t)IB * S * NN;   // [16]
}

// --- 1) firing rates of all 5 source populations -----------------------------
__global__ void prep_rates(const float* __restrict__ t,
                           const float* __restrict__ v,
                           const float* __restrict__ phase,
                           float* __restrict__ rates)
{
    int idx = blockIdx.x * blockDim.x + threadIdx.x;
    if (idx < P * NN) {
        rates[idx] = 1.0f / (1.0f + __expf(-v[idx]));        // sigmoid(v)
    } else if (idx < 5 * NN) {
        int n = idx - P * NN;
        float a = TWO_PI * IN_FREQ * t[0] + phase[n];
        rates[idx] = 0.5f * (1.0f + __sinf(a));              // input population
    }
}

// --- 2) 20 x GEMV via V_WMMA_F32_16X16X4_F32 ---------------------------------
// One wave per (synapse, 128-col block, 512-row chunk). Per k-step of 4 rows:
//   A frag : lane0 <- pre[i..i+1], lane16 <- pre[i+2..i+3], rest 0 (row M=0)
//   B frag k-layout: vgpr0 lanes0-15 = W[i][j], lanes16-31 = W[i+2][j]
//                    vgpr1 lanes0-15 = W[i+1][j], lanes16-31 = W[i+3][j]
// 8 j-tiles share one A fragment: 8 WMMAs per 2 KB of W streamed; the 16 b32
// loads of an iteration cover 4 rows x 512 B contiguous -> full 128 B lines.
__global__ __launch_bounds__(32)
void gemv_wmma(const float* __restrict__ W,      // [S][NN][NN]
               const float* __restrict__ rates,  // [5][NN]
               const int*   __restrict__ src_idx,
               float* __restrict__ drive_part)   // [IB][S][NN]
{
    int bid = blockIdx.x;
    int s   = bid / (JB2 * IB);
    int r   = bid % (JB2 * IB);
    int jb  = r / IB;
    int ic  = r % IB;

    int lane = threadIdx.x;        // wave32
    int half = lane >> 4;          // 0: K=0,1   1: K=2,3
    int lm   = lane & 15;          // column within a 16-wide tile / row M

    const float* pre = rates + (size_t)src_idx[s] * NN + ic * ICH + half * 2;
    const float* Wp  = W + (size_t)s * NN * NN + (size_t)(ic * ICH) * NN
                         + jb * (16 * JT) + lm + half * (2 * NN);

    v8f c[JT];
    #pragma unroll
    for (int k = 0; k < JT; ++k) c[k] = (v8f){};

    const bool row0 = (lm == 0);   // only A-matrix row M=0 carries the vector

    #pragma unroll 2
    for (int it = 0; it < ICH / 4; ++it) {
        v2f pv = *(const v2f*)pre;                 // pre[i+2*half .. +1]
        v2f a;
        a.x = row0 ? pv.x : 0.0f;
        a.y = row0 ? pv.y : 0.0f;

        v2f b[JT];
        #pragma unroll
        for (int k = 0; k < JT; ++k) {             // 4x16 W tile per j-tile
            b[k].x = Wp[16 * k];
            b[k].y = Wp[NN + 16 * k];
        }
        #pragma unroll
        for (int k = 0; k < JT; ++k) {
            c[k] = __builtin_amdgcn_wmma_f32_16x16x4_f32(
                       false, a, false, b[k], (short)0, c[k], false, false);
        }

        Wp  += 4 * NN;
        pre += 4;
    }

    // D row M=0 lives in VGPR 0 (vector element 0) of lanes 0..15.
    if (lane < 16) {
        float* dst = drive_part + ((size_t)ic * S + s) * NN + jb * (16 * JT) + lm;
        #pragma unroll
        for (int k = 0; k < JT; ++k) dst[16 * k] = c[k][0];
    }
}

// --- 3) synapse dynamics, segment-sum, Euler update, outputs -----------------
__global__ __launch_bounds__(256)
void finalize(const float* __restrict__ v,          // [P][NN]
              const float* __restrict__ s_old,      // [S][NN]
              const float* __restrict__ drive_part, // [IB][S][NN]
              const float* __restrict__ E_rev,      // [S]
              const int*   __restrict__ tgt_idx,    // [S]
              float* __restrict__ out,              // [P*NN+1 | P*NN | S*NN]
              float* __restrict__ err_part)         // [gridDim.x]
{
    __shared__ float red[256];
    int tid = threadIdx.x;
    int n   = blockIdx.x * blockDim.x + tid;

    float vloc[P], I[P], G[P];
    #pragma unroll
    for (int p = 0; p < P; ++p) { vloc[p] = v[p * NN + n]; I[p] = 0.f; G[p] = 0.f; }

    float* v_out = out + (P * NN + 1);
    float* s_out = v_out + P * NN;

    #pragma unroll
    for (int sy = 0; sy < S; ++sy) {
        float dr = 0.f;
        #pragma unroll
        for (int ic = 0; ic < IB; ++ic)                      // fixed-order split-K sum
            dr += drive_part[((size_t)ic * S + sy) * NN + n];

        float so = s_old[sy * NN + n];
        float sn = so + DT * (-so * (1.0f / TAU_S) + dr);    // exp. synapse
        s_out[sy * NN + n] = sn;

        int   tg = tgt_idx[sy];
        float er = E_rev[sy];
        #pragma unroll
        for (int p = 0; p < P; ++p) {                        // masked segment-sum
            bool m = (tg == p);
            I[p] += m ? sn * (er - vloc[p]) : 0.f;
            G[p] += m ? sn : 0.f;
        }
    }

    float errloc = 0.f;
    #pragma unroll
    for (int p = 0; p < P; ++p) {
        float dv = (-vloc[p] + I[p] - G[p] * vloc[p]) * (1.0f / TAU_M);
        float vn = vloc[p] + DT * dv;
        errloc += fabsf(dv);
        out[p * NN + n]   = 1.0f / (1.0f + __expf(-vn));     // rates
        v_out[p * NN + n] = vn;
    }

    // deterministic block reduction for the stability error
    red[tid] = errloc;
    __syncthreads();
    #pragma unroll
    for (int off = 128; off > 0; off >>= 1) {
        if (tid < off) red[tid] += red[tid + off];
        __syncthreads();
    }
    if (tid == 0) err_part[blockIdx.x] = red[0];
}

__global__ void write_err(const float* __restrict__ err_part, float* __restrict__ out)
{
    float acc = 0.f;
    #pragma unroll
    for (int b = 0; b < NN / 256; ++b) acc += err_part[b];   // fixed order
    out[P * NN] = acc * (1.0f / (float)NN);                  // sum_p mean|dv|
}

extern "C" void kernel_launch(void* const* d_in, const int* in_sizes, int n_in,
                              void* d_out, int out_size, void* d_ws, size_t ws_size,
                              hipStream_t stream)
{
    const float* t     = (const float*)d_in[0];
    const float* v     = (const float*)d_in[1];
    const float* s     = (const float*)d_in[2];
    const float* W     = (const float*)d_in[3];
    const float* E_rev = (const float*)d_in[4];
    const float* phase = (const float*)d_in[5];
    const int*   src   = (const int*)d_in[6];
    const int*   tgt   = (const int*)d_in[7];
    float*       out   = (float*)d_out;

    float* ws    = (float*)d_ws;
    float* rates = ws + WS_RATES;   // 5*4096
    float* drive = ws + WS_DRIVE;   // 8*20*4096 split-K partials
    float* errp  = ws + WS_ERR;     // 16 block partials

    prep_rates<<<(5 * NN + 255) / 256, 256, 0, stream>>>(t, v, phase, rates);
    gemv_wmma<<<S * JB2 * IB, 32, 0, stream>>>(W, rates, src, drive);
    finalize<<<NN / 256, 256, 0, stream>>>(v, s, drive, E_rev, tgt, out, errp);
    write_err<<<1, 1, 0, stream>>>(errp, out);
}